// AssociativeMemory_88381837017752
// MI455X (gfx1250) — compile-verified
//
#include <hip/hip_runtime.h>
#include <math.h>

#define B_TOT   16384
#define S_SLOTS 32
#define D_DIM   64
#define WPB     4                   // waves per block
#define TILE_B  16                  // batch rows per wave (WMMA M=16)
#define BLK_B   (WPB * TILE_B)      // 64 batches per block
#define NBLK    (B_TOT / BLK_B)     // 256 blocks
#define NWAVE   (B_TOT / TILE_B)    // 1024 waves -> entropy partials

typedef float v2f __attribute__((ext_vector_type(2)));
typedef float v8f __attribute__((ext_vector_type(8)));

__device__ __forceinline__ float wave_max(float v) {
#pragma unroll
  for (int off = 16; off >= 1; off >>= 1) v = fmaxf(v, __shfl_xor(v, off, 32));
  return v;
}
__device__ __forceinline__ float wave_sum(float v) {
#pragma unroll
  for (int off = 16; off >= 1; off >>= 1) v += __shfl_xor(v, off, 32);
  return v;
}
__device__ __forceinline__ int wave_min_i(int v) {
#pragma unroll
  for (int off = 16; off >= 1; off >>= 1) {
    int o = __shfl_xor(v, off, 32);
    v = (o < v) ? o : v;
  }
  return v;
}

__global__ __launch_bounds__(WPB * 32)
void assoc_mem_kernel(const float* __restrict__ gw_r, const float* __restrict__ gw_i,
                      const float* __restrict__ mem_r, const float* __restrict__ mem_i,
                      const float* __restrict__ Wg,   const float* __restrict__ bgp,
                      const float* __restrict__ Wa,   const float* __restrict__ ba,
                      const float* __restrict__ g_r,  const float* __restrict__ b_r,
                      const float* __restrict__ g_i,  const float* __restrict__ b_i,
                      float* __restrict__ read_r, float* __restrict__ read_i,
                      float* __restrict__ next_r, float* __restrict__ next_i,
                      float* __restrict__ ent_ws) {
  // per-wave flat tile [16 rows x 128], stride 132 -> bank (4m+k)&63, conflict-free for m<16
  __shared__ float ldsFlat[WPB][TILE_B][132];
  __shared__ float ldsLogit[WPB][TILE_B][S_SLOTS];
  __shared__ float ldsGate[WPB][TILE_B];

  const int lane = threadIdx.x & 31;
  const int w    = threadIdx.x >> 5;
  const int base = blockIdx.x * BLK_B + w * TILE_B;

  // ---------------- P0: stage flat = [gw_r | gw_i] tile into LDS (coalesced) --------
#pragma unroll 1
  for (int b = 0; b < TILE_B; ++b) {
    const float2 r2 = *(const float2*)(gw_r + (size_t)(base + b) * D_DIM + 2 * lane);
    const float2 i2 = *(const float2*)(gw_i + (size_t)(base + b) * D_DIM + 2 * lane);
    *(float2*)&ldsFlat[w][b][2 * lane]      = r2;
    *(float2*)&ldsFlat[w][b][64 + 2 * lane] = i2;
  }

  // ---------------- P1: logits = flat @ Wa via V_WMMA_F32_16X16X4_F32 ---------------
  {
    const int kh = lane >> 4;        // 0: K 0..1 half, 1: K 2..3 half
    const int n  = lane & 15;        // N within tile / M row for A
    const float* fl = &ldsFlat[w][n][0];   // A: lanes hold row M=n
    v8f acc0 = {};                   // N = 0..15
    v8f acc1 = {};                   // N = 16..31
#pragma unroll
    for (int kc = 0; kc < 32; ++kc) {
      const int kk = kc * 4 + 2 * kh;
      v2f a;  a.x  = fl[kk];                 a.y  = fl[kk + 1];
      v2f b0; b0.x = Wa[kk * 32 + n];        b0.y = Wa[(kk + 1) * 32 + n];
      v2f b1; b1.x = Wa[kk * 32 + 16 + n];   b1.y = Wa[(kk + 1) * 32 + 16 + n];
      acc0 = __builtin_amdgcn_wmma_f32_16x16x4_f32(false, a, false, b0, (short)0, acc0, false, false);
      acc1 = __builtin_amdgcn_wmma_f32_16x16x4_f32(false, a, false, b1, (short)0, acc1, false, false);
    }
    // C layout: VGPR r -> (M = r + 8*kh, N = lane&15)
#pragma unroll
    for (int r = 0; r < 8; ++r) {
      const int M = r + 8 * kh;
      ldsLogit[w][M][n]      = acc0[r] + ba[n];
      ldsLogit[w][M][n + 16] = acc1[r] + ba[n + 16];
    }
  }

  // ---------------- P2: write gate = sigmoid(flat . Wg + bg) ------------------------
#pragma unroll 1
  for (int b = 0; b < TILE_B; ++b) {
    const float* fb = &ldsFlat[w][b][0];
    float p = 0.f;
#pragma unroll
    for (int j = 0; j < 4; ++j) p = fmaf(fb[lane + 32 * j], Wg[lane + 32 * j], p);
    p = wave_sum(p);
    if (lane == 0) ldsGate[w][b] = 1.f / (1.f + __expf(-(p + bgp[0])));
  }

  // ---------------- main loop over the 16 batch rows of this wave -------------------
  float entAcc = 0.f;
#pragma unroll 1
  for (int b = 0; b < TILE_B; ++b) {
    const int bi = base + b;
    const float* fr = &ldsFlat[w][b][0];   // [0..63]=gw_r, [64..127]=gw_i (broadcast reads)
    const float* mrow_r = mem_r + ((size_t)bi * S_SLOTS + lane) * D_DIM;
    const float* mrow_i = mem_i + ((size_t)bi * S_SLOTS + lane) * D_DIM;

    // ---- (a) softmax(logits) -> ww, entropy, top-3 sparse, eu  (lane = slot) ----
    float lg = ldsLogit[w][b][lane];
    float ww;
    {
      const float mx = wave_max(lg);
      const float e  = __expf(lg - mx);
      ww = e / wave_sum(e);
    }
    entAcc -= ww * __logf(ww + 1e-10f);

    bool keep = false;
    {
      float v = ww;
#pragma unroll
      for (int it = 0; it < 3; ++it) {
        const float m2 = wave_max(v);
        const int win  = wave_min_i((v == m2) ? lane : 64);  // lowest lane wins ties
        if (lane == win) { keep = true; v = -1e30f; }
      }
    }
    const float ssum = wave_sum(keep ? ww : 0.f);
    const float eu   = ldsGate[w][b] * (keep ? (ww / (ssum + 1e-6f)) : 0.f);

    // ---- (b) sim + attn  (lane = slot; each lane streams its own row) ----
    float sim = 0.f;
#pragma unroll
    for (int d = 0; d < D_DIM; d += 4) {
      const float4 mr = *(const float4*)(mrow_r + d);
      const float4 mi = *(const float4*)(mrow_i + d);
      sim = fmaf(mr.x, fr[d],          fmaf(mr.y, fr[d + 1],
            fmaf(mr.z, fr[d + 2],     fmaf(mr.w, fr[d + 3], sim))));
      sim = fmaf(mi.x, fr[64 + d],     fmaf(mi.y, fr[64 + d + 1],
            fmaf(mi.z, fr[64 + d + 2], fmaf(mi.w, fr[64 + d + 3], sim))));
    }
    float attn;
    {
      const float mx = wave_max(sim);
      const float e  = __expf(sim - mx);
      attn = e / wave_sum(e);
    }

    // ---- (c) read = attn @ mem  (lane = dim pair; coalesced; attn via shfl bcast) ----
    {
      const float* br = mem_r + (size_t)bi * S_SLOTS * D_DIM + 2 * lane;
      const float* bi_ = mem_i + (size_t)bi * S_SLOTS * D_DIM + 2 * lane;
      float r0 = 0.f, r1 = 0.f, i0 = 0.f, i1 = 0.f;
#pragma unroll
      for (int s = 0; s < S_SLOTS; ++s) {
        const float av = __shfl(attn, s, 32);
        const float2 vr = *(const float2*)(br + s * D_DIM);
        const float2 vi = *(const float2*)(bi_ + s * D_DIM);
        r0 = fmaf(av, vr.x, r0); r1 = fmaf(av, vr.y, r1);
        i0 = fmaf(av, vi.x, i0); i1 = fmaf(av, vi.y, i1);
      }
      float2 orr; orr.x = r0; orr.y = r1;
      float2 oii; oii.x = i0; oii.y = i1;
      *(float2*)(read_r + (size_t)bi * D_DIM + 2 * lane) = orr;
      *(float2*)(read_i + (size_t)bi * D_DIM + 2 * lane) = oii;
    }

    // ---- (d) blend + dual LayerNorm  (lane = slot: row-local stats, no reductions) ----
    {
      const float om = 1.f - eu;
      float s1r = 0.f, s2r = 0.f, s1i = 0.f, s2i = 0.f;
#pragma unroll
      for (int d = 0; d < D_DIM; d += 4) {
        const float4 mr = *(const float4*)(mrow_r + d);
        const float4 mi = *(const float4*)(mrow_i + d);
#pragma unroll
        for (int j = 0; j < 4; ++j) {
          const float mrv = (&mr.x)[j], miv = (&mi.x)[j];
          const float nr = fmaf(om, mrv, eu * fr[d + j]);
          const float ni = fmaf(om, miv, eu * fr[64 + d + j]);
          s1r += nr; s2r = fmaf(nr, nr, s2r);
          s1i += ni; s2i = fmaf(ni, ni, s2i);
        }
      }
      const float mnr = s1r * (1.f / 64.f);
      const float mni = s1i * (1.f / 64.f);
      const float rsr = rsqrtf(fmaf(-mnr, mnr, s2r * (1.f / 64.f)) + 1e-6f);
      const float rsi = rsqrtf(fmaf(-mni, mni, s2i * (1.f / 64.f)) + 1e-6f);
      float* outR = next_r + ((size_t)bi * S_SLOTS + lane) * D_DIM;
      float* outI = next_i + ((size_t)bi * S_SLOTS + lane) * D_DIM;
#pragma unroll
      for (int d = 0; d < D_DIM; d += 4) {
        const float4 mr = *(const float4*)(mrow_r + d);
        const float4 mi = *(const float4*)(mrow_i + d);
        float4 orv, oiv;
#pragma unroll
        for (int j = 0; j < 4; ++j) {
          const float nr = fmaf(om, (&mr.x)[j], eu * fr[d + j]);
          const float ni = fmaf(om, (&mi.x)[j], eu * fr[64 + d + j]);
          (&orv.x)[j] = fmaf((nr - mnr) * rsr, g_r[d + j], b_r[d + j]);
          (&oiv.x)[j] = fmaf((ni - mni) * rsi, g_i[d + j], b_i[d + j]);
        }
        *(float4*)(outR + d) = orv;
        *(float4*)(outI + d) = oiv;
      }
    }
  }

  // deterministic entropy partial: one slot per wave
  const float ent = wave_sum(entAcc);
  if (lane == 0) ent_ws[blockIdx.x * WPB + w] = ent;
}

__global__ __launch_bounds__(256)
void entropy_finalize(const float* __restrict__ ent_ws, float* __restrict__ out_ent) {
  __shared__ float red[256];
  const int t = threadIdx.x;
  float s = 0.f;
  for (int i = t; i < NWAVE; i += 256) s += ent_ws[i];
  red[t] = s;
  __syncthreads();
  for (int st = 128; st >= 1; st >>= 1) {
    if (t < st) red[t] += red[t + st];
    __syncthreads();
  }
  if (t == 0) *out_ent = red[0] * (1.f / (float)B_TOT);
}

extern "C" void kernel_launch(void* const* d_in, const int* in_sizes, int n_in,
                              void* d_out, int out_size, void* d_ws, size_t ws_size,
                              hipStream_t stream) {
  const float* gw_r = (const float*)d_in[0];
  const float* gw_i = (const float*)d_in[1];
  const float* mem_r = (const float*)d_in[2];
  const float* mem_i = (const float*)d_in[3];
  const float* Wg = (const float*)d_in[4];
  const float* bgp = (const float*)d_in[5];
  const float* Wa = (const float*)d_in[6];
  const float* ba = (const float*)d_in[7];
  const float* g_r = (const float*)d_in[8];
  const float* b_r = (const float*)d_in[9];
  const float* g_i = (const float*)d_in[10];
  const float* b_i = (const float*)d_in[11];

  float* out = (float*)d_out;
  float* read_r = out;
  float* read_i = read_r + (size_t)B_TOT * D_DIM;
  float* next_r = read_i + (size_t)B_TOT * D_DIM;
  float* next_i = next_r + (size_t)B_TOT * S_SLOTS * D_DIM;
  float* ent    = next_i + (size_t)B_TOT * S_SLOTS * D_DIM;
  float* ent_ws = (float*)d_ws;   // NWAVE floats

  assoc_mem_kernel<<<NBLK, WPB * 32, 0, stream>>>(
      gw_r, gw_i, mem_r, mem_i, Wg, bgp, Wa, ba, g_r, b_r, g_i, b_i,
      read_r, read_i, next_r, next_i, ent_ws);
  entropy_finalize<<<1, 256, 0, stream>>>(ent_ws, ent);
}